// HGNN_45268955300433
// MI455X (gfx1250) — compile-verified
//
#include <hip/hip_runtime.h>

#define DD 128
#define EPS_GIN 0.1f
#define COEF 0.1f
#define BN_EPS 1e-5f
#define LDSZ 132   // 128 + 4 pad -> stride%64banks = 4, conflict-free
#define LDSH 260   // 256 + 4 pad
#define BN_ROWS 256

typedef __attribute__((ext_vector_type(2))) float v2f;
typedef __attribute__((ext_vector_type(8))) float v8f;

__device__ __forceinline__ float4 ld4(const float* p) { return *(const float4*)p; }

// ---------------- node embedding: h = x_emb1[xa0] + x_emb2[xa1] ----------------
__global__ __launch_bounds__(256) void k_embed(const int* __restrict__ xa,
    const float* __restrict__ emb1, const float* __restrict__ emb2,
    float* __restrict__ h, int N) {
  int tid = blockIdx.x * 256 + threadIdx.x;
  int n = tid >> 5;
  if (n >= N) return;
  int d = (tid & 31) * 4;
  int i1 = xa[n * 2 + 0], i2 = xa[n * 2 + 1];
  float4 a = ld4(emb1 + i1 * DD + d);
  float4 b = ld4(emb2 + i2 * DD + d);
  float4 r; r.x = a.x + b.x; r.y = a.y + b.y; r.z = a.z + b.z; r.w = a.w + b.w;
  *(float4*)(h + n * DD + d) = r;
}

// -------- init agg buffers (fold self-loops + (1+eps)*h_dst analytically) --------
__global__ __launch_bounds__(256) void k_init_dst1(const float* __restrict__ h1,
    const float* __restrict__ e1, const float* __restrict__ e2,
    float* __restrict__ aggZ, float* __restrict__ agg021, int N) {
  int tid = blockIdx.x * 256 + threadIdx.x;
  int n = tid >> 5;
  if (n >= N) return;
  int d = (tid & 31) * 4;
  float4 l1 = ld4(e1 + 4 * DD + d);
  float4 l2 = ld4(e2 + d);
  float4 hv = ld4(h1 + n * DD + d);
  const float c = 2.0f + EPS_GIN;   // self-loop h + (1+eps)*h
  float4 z; z.x = c * hv.x + l1.x + l2.x; z.y = c * hv.y + l1.y + l2.y;
  z.z = c * hv.z + l1.z + l2.z; z.w = c * hv.w + l1.w + l2.w;
  *(float4*)(aggZ + n * DD + d) = z;
  float4 zr = {0.f, 0.f, 0.f, 0.f};
  *(float4*)(agg021 + n * DD + d) = zr;
}

__global__ __launch_bounds__(256) void k_init_dst0(const float* __restrict__ h0,
    const float* __restrict__ e1, const float* __restrict__ e2,
    float* __restrict__ agg030, float* __restrict__ agg110, int N) {
  int tid = blockIdx.x * 256 + threadIdx.x;
  int n = tid >> 5;
  if (n >= N) return;
  int d = (tid & 31) * 4;
  float4 l1 = ld4(e1 + 4 * DD + d);
  float4 l2 = ld4(e2 + d);
  float4 hv = ld4(h0 + n * DD + d);
  float4 z; z.x = hv.x + l1.x + l2.x; z.y = hv.y + l1.y + l2.y;
  z.z = hv.z + l1.z + l2.z; z.w = hv.w + l1.w + l2.w;
  *(float4*)(agg030 + n * DD + d) = z;
  float4 zr = {0.f, 0.f, 0.f, 0.f};
  *(float4*)(agg110 + n * DD + d) = zr;
}

// ---------------- edge scatter: agg[dst] += h_src[src] + e_attr ----------------
__global__ __launch_bounds__(256) void k_scatter(const float* __restrict__ hsrc,
    const int* __restrict__ ei, const int* __restrict__ ea,
    const float* __restrict__ e1, const float* __restrict__ e2,
    float* __restrict__ agg, int E) {
  int tid = blockIdx.x * 256 + threadIdx.x;
  int e = tid >> 5;
  if (e >= E) return;
  int d = (tid & 31) * 4;
  int s  = ei[e];
  int t  = ei[E + e];
  int a0 = ea[e * 2 + 0];
  int a1 = ea[e * 2 + 1];
  float4 hv = ld4(hsrc + s * DD + d);
  float4 v1 = ld4(e1 + a0 * DD + d);
  float4 v2 = ld4(e2 + a1 * DD + d);
  float* p = agg + t * DD + d;
  unsafeAtomicAdd(p + 0, hv.x + v1.x + v2.x);
  unsafeAtomicAdd(p + 1, hv.y + v1.y + v2.y);
  unsafeAtomicAdd(p + 2, hv.z + v1.z + v2.z);
  unsafeAtomicAdd(p + 3, hv.w + v1.w + v2.w);
}

// ---- one 16x16 output tile per wave via V_WMMA_F32_16X16X4_F32, A from LDS ----
__device__ __forceinline__ v8f wave_gemm_lds(const float* lds, int ldstride,
    const float* __restrict__ W, int ncols, int colbase, int K, int lane, v8f acc) {
  const int arow = lane & 15;
  const int koff = (lane < 16) ? 0 : 2;       // A: lanes 0-15 -> K 0,1 ; 16-31 -> K 2,3
  const int col  = colbase + (lane & 15);     // B/C/D: N = lane%16
#pragma unroll 8
  for (int k0 = 0; k0 < K; k0 += 4) {
    v2f a, b;
    const float* zp = lds + arow * ldstride + k0 + koff;
    a.x = zp[0];
    a.y = zp[1];
    b.x = W[(k0 + koff) * ncols + col];
    b.y = W[(k0 + koff + 1) * ncols + col];
    acc = __builtin_amdgcn_wmma_f32_16x16x4_f32(false, a, false, b, (short)0, acc,
                                                false, false);
  }
  return acc;
}

// ---- out1 = 0.5*( relu(z@W1+b1)@W2+b2  +  COEF*(agg021@w021+b021) ), 16 rows/block ----
__global__ __launch_bounds__(256) void k_out1(
    const float* __restrict__ aggZ, const float* __restrict__ aggL,
    const float* __restrict__ w1, const float* __restrict__ b1,
    const float* __restrict__ w2, const float* __restrict__ b2,
    const float* __restrict__ wl, const float* __restrict__ bl,
    float* __restrict__ out, int M) {
  __shared__ float zs[16 * LDSZ];
  __shared__ float hs[16 * LDSH];
  const int tid  = threadIdx.x;
  const int lane = tid & 31, wave = tid >> 5;
  const int row0 = (int)blockIdx.x * 16;

  for (int i = tid; i < 512; i += 256) {          // stage 16x128 z tile
    int m = i >> 5, k4 = (i & 31) * 4;
    int gr = row0 + m; if (gr >= M) gr = M - 1;
    *(float4*)(zs + m * LDSZ + k4) = ld4(aggZ + gr * DD + k4);
  }
  __syncthreads();

#pragma unroll
  for (int tt = 0; tt < 2; ++tt) {                // hidden: 16 col-tiles / 8 waves
    int colbase = (wave + tt * 8) * 16;
    v8f acc = {};
    acc = wave_gemm_lds(zs, LDSZ, w1, 256, colbase, 128, lane, acc);
    float bias = b1[colbase + (lane & 15)];
    int mh = (lane < 16) ? 0 : 8;
#pragma unroll
    for (int r = 0; r < 8; ++r) {
      float h = acc[r] + bias;
      hs[(r + mh) * LDSH + colbase + (lane & 15)] = fmaxf(h, 0.f);
    }
  }
  __syncthreads();

  for (int i = tid; i < 512; i += 256) {          // restage: agg021 tile
    int m = i >> 5, k4 = (i & 31) * 4;
    int gr = row0 + m; if (gr >= M) gr = M - 1;
    *(float4*)(zs + m * LDSZ + k4) = ld4(aggL + gr * DD + k4);
  }
  __syncthreads();

  const int colbase = wave * 16;
  v8f accg = {}; accg = wave_gemm_lds(hs, LDSH, w2, 128, colbase, 256, lane, accg);
  v8f accl = {}; accl = wave_gemm_lds(zs, LDSZ, wl, 128, colbase, 128, lane, accl);
  float bg = b2[colbase + (lane & 15)];
  float bv = bl[colbase + (lane & 15)];
  int mh = (lane < 16) ? 0 : 8;
#pragma unroll
  for (int r = 0; r < 8; ++r) {
    int gr = row0 + r + mh;
    if (gr < M)
      out[gr * DD + colbase + (lane & 15)] =
          0.5f * ((accg[r] + bg) + COEF * (accl[r] + bv));
  }
}

// ---- out0 = 0.5*COEF*( agg110@w110+b110 + agg030@w030+b030 ) ----
__global__ __launch_bounds__(256) void k_out0(
    const float* __restrict__ aggA, const float* __restrict__ aggB,
    const float* __restrict__ wa, const float* __restrict__ ba,
    const float* __restrict__ wb, const float* __restrict__ bb,
    float* __restrict__ out, int M) {
  __shared__ float zs[16 * LDSZ];
  const int tid  = threadIdx.x;
  const int lane = tid & 31, wave = tid >> 5;
  const int row0 = (int)blockIdx.x * 16;
  const int colbase = wave * 16;

  for (int i = tid; i < 512; i += 256) {
    int m = i >> 5, k4 = (i & 31) * 4;
    int gr = row0 + m; if (gr >= M) gr = M - 1;
    *(float4*)(zs + m * LDSZ + k4) = ld4(aggA + gr * DD + k4);
  }
  __syncthreads();
  v8f a1 = {}; a1 = wave_gemm_lds(zs, LDSZ, wa, 128, colbase, 128, lane, a1);
  __syncthreads();
  for (int i = tid; i < 512; i += 256) {
    int m = i >> 5, k4 = (i & 31) * 4;
    int gr = row0 + m; if (gr >= M) gr = M - 1;
    *(float4*)(zs + m * LDSZ + k4) = ld4(aggB + gr * DD + k4);
  }
  __syncthreads();
  v8f a2 = {}; a2 = wave_gemm_lds(zs, LDSZ, wb, 128, colbase, 128, lane, a2);

  float bsum = ba[colbase + (lane & 15)] + bb[colbase + (lane & 15)];
  int mh = (lane < 16) ? 0 : 8;
#pragma unroll
  for (int r = 0; r < 8; ++r) {
    int gr = row0 + r + mh;
    if (gr < M)
      out[gr * DD + colbase + (lane & 15)] = 0.5f * COEF * (a1[r] + a2[r] + bsum);
  }
}

// ---------------- BatchNorm: two-pass batch statistics ----------------
__global__ void k_zero(float* p, int n) {
  int i = blockIdx.x * blockDim.x + threadIdx.x;
  if (i < n) p[i] = 0.f;
}

__global__ __launch_bounds__(128) void k_bnsum(const float* __restrict__ X,
    float* __restrict__ sum, int M) {
  int col = threadIdx.x;
  int r0 = blockIdx.x * BN_ROWS;
  int r1 = r0 + BN_ROWS; if (r1 > M) r1 = M;
  float acc = 0.f;
  for (int r = r0; r < r1; ++r) acc += X[r * DD + col];
  unsafeAtomicAdd(&sum[col], acc);
}

__global__ __launch_bounds__(128) void k_bnvar(const float* __restrict__ X,
    const float* __restrict__ sum, float* __restrict__ vs, int M) {
  int col = threadIdx.x;
  float mu = sum[col] / (float)M;
  int r0 = blockIdx.x * BN_ROWS;
  int r1 = r0 + BN_ROWS; if (r1 > M) r1 = M;
  float acc = 0.f;
  for (int r = r0; r < r1; ++r) { float d = X[r * DD + col] - mu; acc += d * d; }
  unsafeAtomicAdd(&vs[col], acc);
}

__global__ __launch_bounds__(256) void k_bnnorm(const float* __restrict__ X,
    const float* __restrict__ sum, const float* __restrict__ vs,
    const float* __restrict__ gamma, const float* __restrict__ beta,
    float* __restrict__ dst, int M, int doRelu) {
  int tid = blockIdx.x * 256 + threadIdx.x;
  int n = tid >> 5;
  if (n >= M) return;
  int d = (tid & 31) * 4;
  float4 x = ld4(X + n * DD + d);
  float xi[4] = {x.x, x.y, x.z, x.w};
  float o[4];
#pragma unroll
  for (int j = 0; j < 4; ++j) {
    int c = d + j;
    float mu  = sum[c] / (float)M;
    float var = vs[c] / (float)M;
    float y = (xi[j] - mu) * rsqrtf(var + BN_EPS) * gamma[c] + beta[c];
    if (doRelu) y = fmaxf(y, 0.f);
    o[j] = y;
  }
  float4 r = {o[0], o[1], o[2], o[3]};
  *(float4*)(dst + n * DD + d) = r;
}

extern "C" void kernel_launch(void* const* d_in, const int* in_sizes, int n_in,
                              void* d_out, int out_size, void* d_ws, size_t ws_size,
                              hipStream_t stream) {
  (void)n_in; (void)out_size; (void)ws_size;
  const int*   x0     = (const int*)d_in[0];
  const int*   x1     = (const int*)d_in[1];
  const int*   ei101  = (const int*)d_in[2];
  const int*   ea101  = (const int*)d_in[3];
  const int*   ei110  = (const int*)d_in[4];
  const int*   ea110  = (const int*)d_in[5];
  const int*   ei021  = (const int*)d_in[6];
  const int*   ea021  = (const int*)d_in[7];
  const int*   ei030  = (const int*)d_in[8];
  const int*   ea030  = (const int*)d_in[9];
  const float* x_emb1 = (const float*)d_in[10];
  const float* x_emb2 = (const float*)d_in[11];
  const float* e_emb1 = (const float*)d_in[12];
  const float* e_emb2 = (const float*)d_in[13];
  const float* gw1    = (const float*)d_in[14];
  const float* gb1    = (const float*)d_in[15];
  const float* gw2    = (const float*)d_in[16];
  const float* gb2    = (const float*)d_in[17];
  const float* w110   = (const float*)d_in[18];
  const float* b110   = (const float*)d_in[19];
  const float* w021   = (const float*)d_in[20];
  const float* b021   = (const float*)d_in[21];
  const float* w030   = (const float*)d_in[22];
  const float* b030   = (const float*)d_in[23];
  const float* bng    = (const float*)d_in[24];
  const float* bnb    = (const float*)d_in[25];

  const int N0 = in_sizes[0] / 2;
  const int N1 = in_sizes[1] / 2;
  const int E  = in_sizes[3] / 2;

  float* ws = (float*)d_ws;
  size_t off = 0;
  float* h0     = ws + off; off += (size_t)N0 * DD;
  float* h1     = ws + off; off += (size_t)N1 * DD;
  float* aggZ   = ws + off; off += (size_t)N1 * DD;
  float* agg021 = ws + off; off += (size_t)N1 * DD;
  float* agg110 = ws + off; off += (size_t)N0 * DD;
  float* agg030 = ws + off; off += (size_t)N0 * DD;
  float* stats  = ws + off; off += 512;
  float* sum0 = stats, *var0 = stats + 128, *sum1 = stats + 256, *var1 = stats + 384;

  auto blocks = [](long t) { return dim3((unsigned)((t + 255) / 256)); };

  k_embed<<<blocks((long)N0 * 32), 256, 0, stream>>>(x0, x_emb1, x_emb2, h0, N0);
  k_embed<<<blocks((long)N1 * 32), 256, 0, stream>>>(x1, x_emb1, x_emb2, h1, N1);

  float* out0_final = (float*)d_out;
  float* out1_final = (float*)d_out + (size_t)N0 * DD;

  for (int layer = 0; layer < 3; ++layer) {
    k_zero<<<1, 512, 0, stream>>>(stats, 512);
    k_init_dst1<<<blocks((long)N1 * 32), 256, 0, stream>>>(h1, e_emb1, e_emb2, aggZ, agg021, N1);
    k_init_dst0<<<blocks((long)N0 * 32), 256, 0, stream>>>(h0, e_emb1, e_emb2, agg030, agg110, N0);

    dim3 ge = blocks((long)E * 32);
    k_scatter<<<ge, 256, 0, stream>>>(h1, ei101, ea101, e_emb1, e_emb2, aggZ,   E);
    k_scatter<<<ge, 256, 0, stream>>>(h0, ei021, ea021, e_emb1, e_emb2, agg021, E);
    k_scatter<<<ge, 256, 0, stream>>>(h1, ei110, ea110, e_emb1, e_emb2, agg110, E);
    k_scatter<<<ge, 256, 0, stream>>>(h0, ei030, ea030, e_emb1, e_emb2, agg030, E);

    // h0/h1 are dead now -> GEMM outputs overwrite them in place
    k_out1<<<dim3((unsigned)((N1 + 15) / 16)), 256, 0, stream>>>(
        aggZ, agg021, gw1, gb1, gw2, gb2, w021, b021, h1, N1);
    k_out0<<<dim3((unsigned)((N0 + 15) / 16)), 256, 0, stream>>>(
        agg110, agg030, w110, b110, w030, b030, h0, N0);

    dim3 gs0((unsigned)((N0 + BN_ROWS - 1) / BN_ROWS));
    dim3 gs1((unsigned)((N1 + BN_ROWS - 1) / BN_ROWS));
    k_bnsum<<<gs0, 128, 0, stream>>>(h0, sum0, N0);
    k_bnsum<<<gs1, 128, 0, stream>>>(h1, sum1, N1);
    k_bnvar<<<gs0, 128, 0, stream>>>(h0, sum0, var0, N0);
    k_bnvar<<<gs1, 128, 0, stream>>>(h1, sum1, var1, N1);

    const float* g = bng + layer * DD;
    const float* b = bnb + layer * DD;
    int relu = (layer < 2) ? 1 : 0;
    float* d0 = (layer < 2) ? h0 : out0_final;
    float* d1 = (layer < 2) ? h1 : out1_final;
    k_bnnorm<<<blocks((long)N0 * 32), 256, 0, stream>>>(h0, sum0, var0, g, b, d0, N0, relu);
    k_bnnorm<<<blocks((long)N1 * 32), 256, 0, stream>>>(h1, sum1, var1, g, b, d1, N1, relu);
  }
}